// EfficientMemoryHadamard_51969104281833
// MI455X (gfx1250) — compile-verified
//
#include <hip/hip_runtime.h>

// Elementwise Hadamard: out = x1 * x2, f32, 4*4096*4096 = 67,108,864 elements.
// Pure HBM stream (805 MB traffic, ~34.5 us at 23.3 TB/s). Strategy:
//   - b128 (float4) vector loads/stores
//   - non-temporal TH hints (streaming data must not thrash 192 MB L2)
//   - 4 independent b128 loads per input per thread for deep MLP
//   - wave32-friendly 256-thread blocks, fully coalesced block tiling

typedef float v4f __attribute__((ext_vector_type(4)));

#define ELEMS_PER_THREAD 4  // 4 x float4 = 64 B per stream per thread

__global__ __launch_bounds__(256) void hadamard_f32_kernel(
    const v4f* __restrict__ a,
    const v4f* __restrict__ b,
    v4f* __restrict__ out,
    unsigned int n4)  // number of float4 elements
{
    const unsigned int tile = blockDim.x * ELEMS_PER_THREAD;
    unsigned int base = blockIdx.x * tile + threadIdx.x;

    // Fast path: full tile in range -> 4 independent NT b128 loads per stream,
    // coalesced per-instruction (stride = blockDim.x between the 4 accesses).
    if (base + 3u * blockDim.x < n4) {
        const unsigned int i0 = base;
        const unsigned int i1 = base + blockDim.x;
        const unsigned int i2 = base + 2u * blockDim.x;
        const unsigned int i3 = base + 3u * blockDim.x;

        v4f a0 = __builtin_nontemporal_load(a + i0);
        v4f a1 = __builtin_nontemporal_load(a + i1);
        v4f a2 = __builtin_nontemporal_load(a + i2);
        v4f a3 = __builtin_nontemporal_load(a + i3);
        v4f b0 = __builtin_nontemporal_load(b + i0);
        v4f b1 = __builtin_nontemporal_load(b + i1);
        v4f b2 = __builtin_nontemporal_load(b + i2);
        v4f b3 = __builtin_nontemporal_load(b + i3);

        __builtin_nontemporal_store(a0 * b0, out + i0);
        __builtin_nontemporal_store(a1 * b1, out + i1);
        __builtin_nontemporal_store(a2 * b2, out + i2);
        __builtin_nontemporal_store(a3 * b3, out + i3);
    } else {
        // Tail (not taken for 4*4096*4096, but keeps the kernel shape-robust).
        for (int k = 0; k < ELEMS_PER_THREAD; ++k) {
            unsigned int i = base + (unsigned int)k * blockDim.x;
            if (i < n4) {
                v4f av = __builtin_nontemporal_load(a + i);
                v4f bv = __builtin_nontemporal_load(b + i);
                __builtin_nontemporal_store(av * bv, out + i);
            }
        }
    }
}

extern "C" void kernel_launch(void* const* d_in, const int* in_sizes, int n_in,
                              void* d_out, int out_size, void* d_ws, size_t ws_size,
                              hipStream_t stream) {
    (void)n_in; (void)d_ws; (void)ws_size; (void)in_sizes;

    const v4f* a = (const v4f*)d_in[0];
    const v4f* b = (const v4f*)d_in[1];
    v4f* out = (v4f*)d_out;

    // out_size is the flat f32 element count; divisible by 4 for this shape.
    const unsigned int n4 = (unsigned int)(out_size / 4);

    const int threads = 256;                       // 8 wave32 waves per block
    const unsigned int tile = threads * ELEMS_PER_THREAD;  // 1024 float4 per block
    const unsigned int blocks = (n4 + tile - 1u) / tile;   // 16384 blocks at full size

    hadamard_f32_kernel<<<blocks, threads, 0, stream>>>(a, b, out, n4);
}